// EMDLoss_15281493639616
// MI455X (gfx1250) — compile-verified
//
#include <hip/hip_runtime.h>
#include <math.h>

// EMD interior-point solver for MI455X (gfx1250), wave32.
// One batch = one wave. Woodbury-structured KKT solve; the only dense
// matmul (16x16x16 fp32 syrk for the Schur complement) uses
// V_WMMA_F32_16X16X4_F32.

typedef float v8f __attribute__((ext_vector_type(8)));
typedef float v2f __attribute__((ext_vector_type(2)));

#define L2REG  1.0e-4f
#define SIGW   0.1f
#define NITER  20

__device__ __forceinline__ float wsum32(float v) {
  v += __shfl_xor(v, 1, 32);
  v += __shfl_xor(v, 2, 32);
  v += __shfl_xor(v, 4, 32);
  v += __shfl_xor(v, 8, 32);
  v += __shfl_xor(v, 16, 32);
  return v;
}
__device__ __forceinline__ float wsum16(float v) {
  v += __shfl_xor(v, 1, 32);
  v += __shfl_xor(v, 2, 32);
  v += __shfl_xor(v, 4, 32);
  v += __shfl_xor(v, 8, 32);
  return v;
}
__device__ __forceinline__ float wmin32(float v) {
  v = fminf(v, __shfl_xor(v, 1, 32));
  v = fminf(v, __shfl_xor(v, 2, 32));
  v = fminf(v, __shfl_xor(v, 4, 32));
  v = fminf(v, __shfl_xor(v, 8, 32));
  v = fminf(v, __shfl_xor(v, 16, 32));
  return v;
}

// Element mapping: flat k = i*16 + j lives at register r = k/32, lane = k%32,
// with i = 2r + (lane>>4), j = lane&15.  16-vectors are "lane-vectors":
// value for index t at lanes t and t+16 (replicated halves).
__global__ __launch_bounds__(32) void emd_ip_wmma_kernel(
    const float* __restrict__ jets1,
    const float* __restrict__ jets2,
    float* __restrict__ out) {
  __shared__ float Ml[16 * 17];   // M = Ka^{-1/2} E, staged for WMMA operands
  __shared__ float Sl[16 * 17];   // Sb staged for row-major pickup
  __shared__ float rs16[16];      // row-sum gather
  __shared__ float cs16[16];      // col-sum gather
  __shared__ float bufZ[32];      // z[256..287]
  __shared__ float bufS[32];      // s[256..287]
  __shared__ float bufQ[32];      // q[256..287]

  const int lane = threadIdx.x;
  const int half = lane >> 4;
  const int jc   = lane & 15;
  const int b    = blockIdx.x;

  const float* J1 = jets1 + b * 48;
  const float* J2 = jets2 + b * 48;
  float a1x = J1[jc * 3 + 0], a1y = J1[jc * 3 + 1], w1 = J1[jc * 3 + 2];
  float a2x = J2[jc * 3 + 0], a2y = J2[jc * 3 + 1], w2 = J2[jc * 3 + 2];

  float w1s = wsum16(w1);
  float w2s = wsum16(w2);
  float bsc   = fminf(w1s, w2s);
  float ediff = fabsf(w1s - w2s);

  // State: x (256) -> 8 regs, s/z (288) -> 9 regs, p (256), h tail (32).
  float p8[8], x8[8], s9[9], z9[9];
#pragma unroll
  for (int r = 0; r < 8; ++r) {
    int ir = 2 * r + half;
    float ddx = -(__shfl(a1x, ir, 32) - a2x) + 1e-12f;
    float ddy = -(__shfl(a1y, ir, 32) - a2y) + 1e-12f;
    p8[r] = sqrtf(ddx * ddx + ddy * ddy);
    x8[r] = 0.f; s9[r] = 1.f; z9[r] = 1.f;
  }
  s9[8] = 1.f; z9[8] = 1.f;
  float h8 = (half == 0) ? w1 : w2;   // h[256+i]=w1[i], h[272+j]=w2[j]
  float yv = 0.f;

  for (int it = 0; it < NITER; ++it) {
    // mu = mean(s*z)
    float acc = 0.f;
#pragma unroll
    for (int r = 0; r < 8; ++r) acc += s9[r] * z9[r];
    acc += s9[8] * z9[8];
    float mu = wsum32(acc) * (1.f / 288.f);

    // ra = sum(x) - b
    float tx = 0.f;
#pragma unroll
    for (int r = 0; r < 8; ++r) tx += x8[r];
    float ra = wsum32(tx) - bsc;

    // row/col sums of x (for rp tail)
    __syncthreads();
#pragma unroll
    for (int r = 0; r < 8; ++r) {
      float t = wsum16(x8[r]);
      if (jc == 0) rs16[2 * r + half] = t;
    }
    {
      float c0 = 0.f;
#pragma unroll
      for (int r = 0; r < 8; ++r) c0 += x8[r];
      c0 += __shfl_xor(c0, 16, 32);
      if (lane < 16) cs16[lane] = c0;
    }
    __syncthreads();
    float xr = rs16[jc], xc = cs16[jc];

    // rp = x G^T + s - h ;  rs = s*z - sigma*mu
    float rpv[8], rsv[8];
#pragma unroll
    for (int r = 0; r < 8; ++r) {
      rpv[r] = -x8[r] + s9[r];              // h[k<256] = 0
      rsv[r] = s9[r] * z9[r] - SIGW * mu;
    }
    float rp8 = ((half == 0) ? xr : xc) + s9[8] - h8;
    float rs8 = s9[8] * z9[8] - SIGW * mu;

    // stage tail (c in [256,288)) values for broadcast
    float q8 = (z9[8] * rp8 - rs8) / s9[8];
    __syncthreads();
    bufZ[lane] = z9[8];
    bufS[lane] = s9[8];
    bufQ[lane] = q8;
    __syncthreads();

    // r1 = rx + q G  (k<256 part), Dinv = 1/(L2 + z/s)
    float r1v[8], Di[8];
#pragma unroll
    for (int r = 0; r < 8; ++r) {
      int ir = 2 * r + half;
      float rx = L2REG * x8[r] + p8[r] + (-z9[r] + bufZ[ir] + bufZ[16 + jc]) + yv;
      float q  = (z9[r] * rpv[r] - rsv[r]) / s9[r];
      r1v[r] = rx + (-q + bufQ[ir] + bufQ[16 + jc]);
      Di[r]  = 1.f / (L2REG + z9[r] / s9[r]);
    }

    // E row/col sums -> Ka, Kb diagonals
    __syncthreads();
#pragma unroll
    for (int r = 0; r < 8; ++r) {
      float t = wsum16(Di[r]);
      if (jc == 0) rs16[2 * r + half] = t;
    }
    {
      float c0 = 0.f;
#pragma unroll
      for (int r = 0; r < 8; ++r) c0 += Di[r];
      c0 += __shfl_xor(c0, 16, 32);
      if (lane < 16) cs16[lane] = c0;
    }
    __syncthreads();
    float er = rs16[jc], ec = cs16[jc];

    float ka   = bufS[jc] / bufZ[jc] + er;             // 1/dA + rowsum(E)
    float kb   = bufS[16 + jc] / bufZ[16 + jc] + ec;   // 1/dB + colsum(E)
    float kain = 1.f / ka;
    float kais = sqrtf(kain);

    // M = Ka^{-1/2} E  -> LDS (row-major, stride 17)
#pragma unroll
    for (int r = 0; r < 8; ++r) {
      int ir = 2 * r + half;
      Ml[ir * 17 + jc] = Di[r] * __shfl(kais, ir, 32);
    }
    __syncthreads();

    // --- WMMA: P = M^T M, fp32 16x16, K=16 as 4 chunks of 4.
    // Since P is a syrk, the A (M^T, 16x4) and B (M, 4x16) operands have
    // identical lane layouts: lane l<16 -> {M[4c][l], M[4c+1][l]},
    // lane l>=16 -> {M[4c+2][l-16], M[4c+3][l-16]}.
    v8f accv = {0.f, 0.f, 0.f, 0.f, 0.f, 0.f, 0.f, 0.f};
#pragma unroll
    for (int c4 = 0; c4 < 4; ++c4) {
      int rowb = 4 * c4 + half * 2;
      v2f op;
      op.x = Ml[rowb * 17 + jc];
      op.y = Ml[(rowb + 1) * 17 + jc];
      accv = __builtin_amdgcn_wmma_f32_16x16x4_f32(
          false, op, false, op, (short)0, accv, false, false);
    }

    // Sb = diag(Kb) - P  -> LDS (D layout: vgpr rr, lanes<16 row rr, else 8+rr)
#pragma unroll
    for (int rr = 0; rr < 8; ++rr) {
      int m = rr + 8 * half;
      float val = -accv[rr];
      if (m == jc) val += kb;
      Sl[m * 17 + jc] = val;
    }
    __syncthreads();
    float Srow[16];
#pragma unroll
    for (int c = 0; c < 16; ++c) Srow[c] = Sl[jc * 17 + c];

    // In-register 16x16 Cholesky: lane jc owns row jc (replicated halves).
    float Lch[16];
#pragma unroll
    for (int k = 0; k < 16; ++k) {
      float skk  = __shfl(Srow[k], k, 32);
      float rinv = rsqrtf(skk);
      float lrk  = Srow[k] * rinv;   // L[r][k], valid for r>=k
      Lch[k] = lrk;
#pragma unroll
      for (int c = k + 1; c < 16; ++c)
        Srow[c] -= lrk * __shfl(lrk, c, 32);
    }

    // Capacity-system solve for one rhs t0 = D^{-1} r  -> (ta, tb) lane-vectors
    auto ksolve = [&](const float (&t0)[8], float& taO, float& tbO) {
      __syncthreads();
#pragma unroll
      for (int r = 0; r < 8; ++r) {
        float t = wsum16(t0[r]);
        if (jc == 0) rs16[2 * r + half] = t;
      }
      {
        float c0 = 0.f;
#pragma unroll
        for (int r = 0; r < 8; ++r) c0 += t0[r];
        c0 += __shfl_xor(c0, 16, 32);
        if (lane < 16) cs16[lane] = c0;
      }
      __syncthreads();
      float ga = rs16[jc], gb = cs16[jc];
      // gb' = gb - E^T (ga/Ka)
      float av = ga * kain;
      float s2 = 0.f;
#pragma unroll
      for (int r = 0; r < 8; ++r) s2 += Di[r] * __shfl(av, 2 * r + half, 32);
      s2 += __shfl_xor(s2, 16, 32);
      float greg = gb - s2;
      // forward: L w = gb'
      float wv = 0.f;
#pragma unroll
      for (int k = 0; k < 16; ++k) {
        float lkk = __shfl(Lch[k], k, 32);
        float wk  = __shfl(greg, k, 32) / lkk;
        greg -= Lch[k] * wk;
        wv = (jc == k) ? wk : wv;
      }
      // backward: L^T tb = w
      float tv = 0.f;
#pragma unroll
      for (int k = 15; k >= 0; --k) {
        float prod = wsum16(Lch[k] * tv);   // sum_{c>k} L[c][k] tb[c]
        float lkk  = __shfl(Lch[k], k, 32);
        float tk   = (__shfl(wv, k, 32) - prod) / lkk;
        tv = (jc == k) ? tk : tv;
      }
      tbO = tv;
      // ta = (ga - E tb)/Ka
      __syncthreads();
#pragma unroll
      for (int r = 0; r < 8; ++r) {
        float t = wsum16(Di[r] * tv);
        if (jc == 0) rs16[2 * r + half] = t;
      }
      __syncthreads();
      taO = (ga - rs16[jc]) * kain;
    };

    float t0u[8], t0v[8];
#pragma unroll
    for (int r = 0; r < 8; ++r) { t0u[r] = Di[r] * r1v[r]; t0v[r] = Di[r]; }
    float tau, tbu, tav, tbv;
    ksolve(t0u, tau, tbu);   // u = H^{-1} r1
    ksolve(t0v, tav, tbv);   // v = H^{-1} 1

    float uv[8], vvv[8], su = 0.f, sv = 0.f;
#pragma unroll
    for (int r = 0; r < 8; ++r) {
      int ir = 2 * r + half;
      uv[r]  = Di[r] * (r1v[r] - __shfl(tau, ir, 32) - tbu);
      vvv[r] = Di[r] * (1.f    - __shfl(tav, ir, 32) - tbv);
      su += uv[r]; sv += vvv[r];
    }
    su = wsum32(su); sv = wsum32(sv);
    float dyv = (ra - su) / sv;

    float dxv[8];
#pragma unroll
    for (int r = 0; r < 8; ++r) dxv[r] = -uv[r] - vvv[r] * dyv;

    // row/col sums of dx for ds tail
    __syncthreads();
#pragma unroll
    for (int r = 0; r < 8; ++r) {
      float t = wsum16(dxv[r]);
      if (jc == 0) rs16[2 * r + half] = t;
    }
    {
      float c0 = 0.f;
#pragma unroll
      for (int r = 0; r < 8; ++r) c0 += dxv[r];
      c0 += __shfl_xor(c0, 16, 32);
      if (lane < 16) cs16[lane] = c0;
    }
    __syncthreads();
    float dxr = rs16[jc], dxc = cs16[jc];

    float dsv[8], dzv[8];
    float am = 3.0e38f;
#pragma unroll
    for (int r = 0; r < 8; ++r) {
      dsv[r] = -rpv[r] + dxv[r];
      dzv[r] = (-rsv[r] - z9[r] * dsv[r]) / s9[r];
      if (dsv[r] < 0.f) am = fminf(am, -s9[r] / dsv[r]);
      if (dzv[r] < 0.f) am = fminf(am, -z9[r] / dzv[r]);
    }
    float ds8 = -rp8 - ((half == 0) ? dxr : dxc);
    float dz8 = (-rs8 - z9[8] * ds8) / s9[8];
    if (ds8 < 0.f) am = fminf(am, -s9[8] / ds8);
    if (dz8 < 0.f) am = fminf(am, -z9[8] / dz8);
    am = wmin32(am);
    float alpha = 0.99f * fminf(1.f, am);

#pragma unroll
    for (int r = 0; r < 8; ++r) {
      x8[r] += alpha * dxv[r];
      s9[r] += alpha * dsv[r];
      z9[r] += alpha * dzv[r];
    }
    s9[8] += alpha * ds8;
    z9[8] += alpha * dz8;
    yv += alpha * dyv;
  }

  // emd = sum(p * flow) + |sum w1 - sum w2|
  float e = 0.f;
#pragma unroll
  for (int r = 0; r < 8; ++r) e += p8[r] * x8[r];
  e = wsum32(e) + ediff;
  if (lane == 0) out[b] = e;
}

extern "C" void kernel_launch(void* const* d_in, const int* in_sizes, int n_in,
                              void* d_out, int out_size, void* d_ws, size_t ws_size,
                              hipStream_t stream) {
  (void)n_in; (void)d_ws; (void)ws_size; (void)out_size;
  const float* jets1 = (const float*)d_in[0];
  const float* jets2 = (const float*)d_in[1];
  float* out = (float*)d_out;
  const int B = in_sizes[0] / 48;   // (B, 16, 3) float32
  emd_ip_wmma_kernel<<<dim3(B), dim3(32), 0, stream>>>(jets1, jets2, out);
}